// Attention_16776142258254
// MI455X (gfx1250) — compile-verified
//
#include <hip/hip_runtime.h>
#include <math.h>

// Attention pooling, flash-style single pass over enc_out (1 GiB read once).
// B=64, T=4096, H=1024, fp32. Memory-bound: ~1 GiB / 23.3 TB/s ~= 46 us floor.

typedef float v2f __attribute__((ext_vector_type(2)));
typedef float v8f __attribute__((ext_vector_type(8)));

#define B_      64
#define T_      4096
#define H_      1024
#define SCALE_  0.03125f        // 1/sqrt(1024)
#define CCHUNK  8               // chunks per batch -> 512 workgroups
#define TC_     (T_ / CCHUNK)   // 512 rows per chunk
#define RTILE   32              // rows per LDS tile
#define NTILES  (TC_ / RTILE)   // 16 tiles per chunk
#define LDSTR   1028            // row stride in floats (pad 4 -> conflict-free b64 A loads)

// Dynamic LDS bytes: tile + dec vector + score reduction buffer
#define SMEM_FLOATS (RTILE * LDSTR + H_ + RTILE)
#define SMEM_BYTES  (SMEM_FLOATS * 4)

__device__ __forceinline__ void wait_asynccnt0() {
#if __has_builtin(__builtin_amdgcn_s_wait_asynccnt)
    __builtin_amdgcn_s_wait_asynccnt(0);
#else
    asm volatile("s_wait_asynccnt 0x0" ::: "memory");
#endif
}

__launch_bounds__(256)
__global__ void attn_flash_k1(const float* __restrict__ dec_h,
                              const float* __restrict__ enc,
                              float* __restrict__ scores,     // [B,T] scaled raw scores (weights region of d_out)
                              float* __restrict__ chunk_m,    // [B*CCHUNK]
                              float* __restrict__ chunk_l,    // [B*CCHUNK]
                              float* __restrict__ chunk_ctx)  // [B*CCHUNK, H]
{
    extern __shared__ float smem[];
    float* lds_tile = smem;                      // RTILE * LDSTR
    float* lds_dec  = smem + RTILE * LDSTR;      // H_
    float* s_red    = lds_dec + H_;              // RTILE

    const int tid  = threadIdx.x;
    const int lane = tid & 31;
    const int wave = tid >> 5;
    const int b    = blockIdx.y;
    const int c    = blockIdx.x;
    const int t0   = c * TC_;
    const int h0   = tid * 4;                    // each thread owns 4 h-columns

    // dec_h[b] -> LDS (4 KB)
    {
        const float4 dv = *(const float4*)(dec_h + (size_t)b * H_ + h0);
        *(float4*)(lds_dec + h0) = dv;
    }

    // WMMA work split: 2 row-groups x 4 k-ranges across 8 waves
    const int rg   = wave & 1;                   // rows rg*16 .. rg*16+15
    const int kr   = wave >> 1;                  // k in [kr*256, kr*256+256)
    const int arow = rg * 16 + (lane & 15);      // A-matrix M index for this lane
    const int ksub = (lane >> 4) << 1;           // lanes 0-15: K0/K1, lanes 16-31: K2/K3
    const bool bld = (lane & 15) == 0;           // lanes 0 and 16 hold D column N=0

    float  m_run = -1e30f;
    float  l_run = 0.f;
    float4 ctx; ctx.x = 0.f; ctx.y = 0.f; ctx.z = 0.f; ctx.w = 0.f;

    for (int tile = 0; tile < NTILES; ++tile) {
        const int tbase = t0 + tile * RTILE;

        if (tid < RTILE) s_red[tid] = 0.f;

        // ---- tile fill: async global -> LDS (per-lane 16 B, ASYNCcnt-tracked).
        // Row r (4 KB) is covered by the 256 threads' float4 slots.
        {
            const float* gsrc = enc + ((size_t)b * T_ + tbase) * H_ + h0;
            #pragma unroll 4
            for (int r = 0; r < RTILE; ++r) {
                const uint64_t ga = (uint64_t)(uintptr_t)(gsrc + (size_t)r * H_);
                const uint32_t lo = (uint32_t)(uintptr_t)(lds_tile + r * LDSTR + h0);
                asm volatile("global_load_async_to_lds_b128 %0, %1, off"
                             :: "v"(lo), "v"(ga) : "memory");
            }
        }
        // prefetch next tile (global_prefetch_b8): 2 x 256B-stride covers 128 KB
        if (tile + 1 < NTILES) {
            const char* pn = (const char*)(enc + ((size_t)b * T_ + tbase + RTILE) * H_);
            __builtin_prefetch(pn + tid * 256, 0, 1);
            __builtin_prefetch(pn + 65536 + tid * 256, 0, 1);
        }
        wait_asynccnt0();
        __syncthreads();

        // ---- score pass: D(16x16) = A(16x4) x B(4x16).
        // All lanes broadcast-load the same dec chunk (lanes0-15: K0/K1, 16-31: K2/K3),
        // so every D column holds the same dot products; we read column 0 only.
        v8f acc = {0.f, 0.f, 0.f, 0.f, 0.f, 0.f, 0.f, 0.f};
        const float* aptr = lds_tile + arow * LDSTR + ksub;
        const float* bptr = lds_dec + ksub;
        #pragma unroll 4
        for (int j = 0; j < 64; ++j) {
            const int k = (kr << 8) + (j << 2);
            const v2f av = *(const v2f*)(aptr + k);
            const v2f bv = *(const v2f*)(bptr + k);
            acc = __builtin_amdgcn_wmma_f32_16x16x4_f32(
                false, av, false, bv, (short)0, acc, false, false);
        }
        // column N=0 lives in lane 0 (M = rg*16+0..7) and lane 16 (M = rg*16+8..15)
        if (bld) {
            const int rbase = rg * 16 + (lane >> 4) * 8;
            #pragma unroll
            for (int i = 0; i < 8; ++i)
                atomicAdd(&s_red[rbase + i], acc[i]);
        }
        __syncthreads();

        // ---- online softmax update + ctx accumulation (tile reused from LDS)
        float tmax = -1e30f;
        #pragma unroll
        for (int r = 0; r < RTILE; ++r) tmax = fmaxf(tmax, s_red[r]);
        tmax *= SCALE_;                                   // SCALE>0: commutes with max
        const float m_new = fmaxf(m_run, tmax);
        const float alpha = __expf(m_run - m_new);
        l_run *= alpha;
        ctx.x *= alpha; ctx.y *= alpha; ctx.z *= alpha; ctx.w *= alpha;
        float sump = 0.f;
        #pragma unroll 4
        for (int r = 0; r < RTILE; ++r) {
            const float p = __expf(s_red[r] * SCALE_ - m_new);
            sump += p;
            const float4 tv = *(const float4*)(lds_tile + r * LDSTR + h0);
            ctx.x += p * tv.x; ctx.y += p * tv.y;
            ctx.z += p * tv.z; ctx.w += p * tv.w;
        }
        l_run += sump;
        m_run  = m_new;

        if (tid < RTILE)
            scores[(size_t)b * T_ + tbase + tid] = s_red[tid] * SCALE_;
        __syncthreads();
    }

    const int cid = b * CCHUNK + c;
    if (tid == 0) { chunk_m[cid] = m_run; chunk_l[cid] = l_run; }
    *(float4*)(chunk_ctx + (size_t)cid * H_ + h0) = ctx;
}

__launch_bounds__(256)
__global__ void attn_flash_k2(const float* __restrict__ chunk_m,
                              const float* __restrict__ chunk_l,
                              const float* __restrict__ chunk_ctx,
                              float* __restrict__ ctx_out,   // d_out[0 .. B*H)
                              float* __restrict__ weights)   // d_out[B*H ..): in = scaled scores, out = weights
{
    const int b   = blockIdx.x;
    const int tid = threadIdx.x;

    float mc[CCHUNK], lc[CCHUNK];
    float mg = -1e30f;
    #pragma unroll
    for (int c = 0; c < CCHUNK; ++c) {
        mc[c] = chunk_m[b * CCHUNK + c];
        lc[c] = chunk_l[b * CCHUNK + c];
        mg = fmaxf(mg, mc[c]);
    }
    float lg = 0.f;
    #pragma unroll
    for (int c = 0; c < CCHUNK; ++c) lg += lc[c] * __expf(mc[c] - mg);
    const float inv = 1.f / lg;

    float4 a; a.x = 0.f; a.y = 0.f; a.z = 0.f; a.w = 0.f;
    #pragma unroll
    for (int c = 0; c < CCHUNK; ++c) {
        const float s = __expf(mc[c] - mg);
        const float4 v = *(const float4*)(chunk_ctx + ((size_t)(b * CCHUNK + c)) * H_ + tid * 4);
        a.x += s * v.x; a.y += s * v.y; a.z += s * v.z; a.w += s * v.w;
    }
    a.x *= inv; a.y *= inv; a.z *= inv; a.w *= inv;
    *(float4*)(ctx_out + (size_t)b * H_ + tid * 4) = a;

    for (int t = tid; t < T_; t += 256) {
        const size_t idx = (size_t)b * T_ + t;
        weights[idx] = __expf(weights[idx] - mg) * inv;
    }
}

extern "C" void kernel_launch(void* const* d_in, const int* in_sizes, int n_in,
                              void* d_out, int out_size, void* d_ws, size_t ws_size,
                              hipStream_t stream) {
    const float* dec_h = (const float*)d_in[0];   // [B, H]
    const float* enc   = (const float*)d_in[1];   // [B, T, H]
    float* ctx_out = (float*)d_out;                       // [B, H]
    float* weights = (float*)d_out + (size_t)B_ * H_;     // [B, T]

    // workspace layout (needs ~2.01 MB): m[512] | l[512] | ctx[512*1024]
    float* chunk_m   = (float*)d_ws;
    float* chunk_l   = chunk_m + B_ * CCHUNK;
    float* chunk_ctx = chunk_l + B_ * CCHUNK;

    attn_flash_k1<<<dim3(CCHUNK, B_), 256, SMEM_BYTES, stream>>>(
        dec_h, enc, weights, chunk_m, chunk_l, chunk_ctx);
    attn_flash_k2<<<dim3(B_), 256, 0, stream>>>(
        chunk_m, chunk_l, chunk_ctx, ctx_out, weights);
}